// NetVLAD_44693429682206
// MI455X (gfx1250) — compile-verified
//
#include <hip/hip_runtime.h>
#include <math.h>

typedef __attribute__((ext_vector_type(2))) float v2f;
typedef __attribute__((ext_vector_type(4))) float v4f;
typedef __attribute__((ext_vector_type(8))) float v8f;

#if defined(__has_builtin)
#  if __has_builtin(__builtin_amdgcn_global_load_async_to_lds_b128)
#    define HAVE_ASYNC_LDS 1
#  endif
#endif
#ifndef HAVE_ASYNC_LDS
#  define HAVE_ASYNC_LDS 0
#endif

#define NN 64      // batch
#define CC 128     // channels
#define PP 4096    // pixels per image (H*W)
#define KK 64      // clusters
#define PT 128     // pixel sub-tile
#define NSUB 8     // sub-tiles per chunk
#define NCHUNK 4   // chunks per image -> 256 blocks total
// LDS row strides (floats)
#define SW 132     // conv_w [K][C]
#define SX 136     // xn tile [C][PT] (544B rows, 16B aligned)
#define SK 72      // soft-assign [PT][K] transposed (288B rows, 16B aligned)

#if HAVE_ASYNC_LDS
// builtin signature (from hipcc diagnostic): (int4 AS1*, int4 AS3*, imm offset, imm cpol)
typedef int v4i __attribute__((vector_size(16)));
typedef __attribute__((address_space(1))) v4i GV4;
typedef __attribute__((address_space(3))) v4i LV4;
#endif

__device__ __forceinline__ void wait_async() {
#if HAVE_ASYNC_LDS
#  if __has_builtin(__builtin_amdgcn_s_wait_asynccnt)
    __builtin_amdgcn_s_wait_asynccnt(0);
#  else
    asm volatile("s_wait_asynccnt 0" ::: "memory");
#  endif
#endif
}

// V_WMMA_F32_16X16X4_F32: D[16x16] = A[16x4] * B[4x16] + C, wave32.
// A layout (ISA 7.12.2): lanes 0-15 -> M=lane, {v0,v1} = K{0,1}; lanes 16-31 -> M=lane-16, K{2,3}.
// B layout mirrored (lane = N). C/D: vgpr i -> M = i + 8*(lane>=16), N = lane&15.
__device__ __forceinline__ v8f wmma_f32(v2f a, v2f b, v8f c) {
    return __builtin_amdgcn_wmma_f32_16x16x4_f32(false, a, false, b, (short)0, c, false, false);
}

__global__ __launch_bounds__(256)
void netvlad_main(const float* __restrict__ x,        // [N,C,P]
                  const float* __restrict__ conv_w,   // [K,C]
                  const float* __restrict__ conv_b,   // [K]
                  float* __restrict__ vlad_acc,       // [N,K,C] (zeroed)
                  float* __restrict__ asum_acc)       // [N,K]   (zeroed)
{
    __shared__ float sW[KK * SW];       // conv_w staged
    __shared__ float sX0[CC * SX];      // xn tile, double-buffered
    __shared__ float sX1[CC * SX];
    __shared__ float sA[PT * SK];       // logits -> soft assignment, TRANSPOSED [P][K]
    __shared__ float sAsum[KK];         // running sum_p a[k,p]
    __shared__ float sRed[8 * PT];      // norm partials / inv-norms

    const int tid   = threadIdx.x;
    const int blk   = blockIdx.x;
    const int n     = blk / NCHUNK;
    const int chunk = blk % NCHUNK;

    for (int i = tid; i < KK * CC; i += 256)
        sW[(i >> 7) * SW + (i & 127)] = conv_w[i];
    if (tid < KK) sAsum[tid] = 0.0f;

    const int wv   = tid >> 5;
    const int lane = tid & 31;
    const int lrow = lane & 15;
    const int lkk  = (lane >> 4) << 1;        // K offset within 4-wide inner dim
    const int km   = wv >> 1;                 // 16-row block of K owned by this wave
    const int qn0  = (wv & 1) * 4;            // group of 4 16-col tiles
    const int krow = km * 16 + (lane >> 4) * 8;

    float br[8];                              // bias, register-resident
    #pragma unroll
    for (int i = 0; i < 8; ++i) br[i] = conv_b[krow + i];

    v8f vacc[4] = {};                         // persistent vlad accumulators [K16 x C16] x4

    const int u  = tid & 31;                  // pixel quad (pixels 4u..4u+3)
    const int cg = tid >> 5;                  // channel group (16 channels)

    // async (or sync-fallback) stage of one 128-pixel x tile into an LDS buffer
    const float* gbase = x + (size_t)n * CC * PP + chunk * (NSUB * PT) + 4 * u;
    auto issue_tile = [&](int st2, float* buf) {
        const float* gp = gbase + st2 * PT;
        #pragma unroll
        for (int cc = 0; cc < 16; ++cc) {
            const int c = cg * 16 + cc;
#if HAVE_ASYNC_LDS
            __builtin_amdgcn_global_load_async_to_lds_b128(
                (GV4*)(gp + (size_t)c * PP),
                (LV4*)&buf[c * SX + 4 * u], 0, 0);
#else
            *(v4f*)&buf[c * SX + 4 * u] = *(const v4f*)(gp + (size_t)c * PP);
#endif
        }
    };

    issue_tile(0, sX0);                       // prologue prefetch

    for (int st = 0; st < NSUB; ++st) {
        float* bx = (st & 1) ? sX1 : sX0;

        wait_async();                         // own tile landed in LDS
        __syncthreads();                      // everyone's tile visible; prev iter done
        if (st + 1 < NSUB)                    // prefetch next tile into other buffer
            issue_tile(st + 1, (st & 1) ? sX0 : sX1);

        // ---- channel-wise L2 norm, register-resident ----
        {
            v4f r[16];
            v4f ss4 = {};
            #pragma unroll
            for (int cc = 0; cc < 16; ++cc) {
                const int c = cg * 16 + cc;
                r[cc] = *(const v4f*)&bx[c * SX + 4 * u];
                ss4 += r[cc] * r[cc];
            }
            *(v4f*)&sRed[cg * PT + 4 * u] = ss4;
            __syncthreads();
            if (tid < PT) {
                float s = 0.0f;
                #pragma unroll
                for (int g = 0; g < 8; ++g) s += sRed[g * PT + tid];
                sRed[tid] = 1.0f / fmaxf(sqrtf(s), 1e-12f);
            }
            __syncthreads();
            const v4f inv4 = *(const v4f*)&sRed[4 * u];
            #pragma unroll
            for (int cc = 0; cc < 16; ++cc) {
                const int c = cg * 16 + cc;
                *(v4f*)&bx[c * SX + 4 * u] = r[cc] * inv4;
            }
        }
        __syncthreads();

        // ---- GEMM1: logits[k][p] = sum_c W[k][c] * xn[c][p] ----
        v8f acc[4] = {};
        for (int c0 = 0; c0 < CC; c0 += 4) {
            v2f a;
            const float* wp = &sW[(km * 16 + lrow) * SW + c0 + lkk];
            a.x = wp[0]; a.y = wp[1];
            #pragma unroll
            for (int q = 0; q < 4; ++q) {
                const int col = (qn0 + q) * 16 + lrow;
                v2f b;
                b.x = bx[(c0 + lkk) * SX + col];
                b.y = bx[(c0 + lkk + 1) * SX + col];
                acc[q] = wmma_f32(a, b, acc[q]);
            }
        }
        // bias + store logits transposed: lane's 8 D values contiguous -> 2x b128
        #pragma unroll
        for (int q = 0; q < 4; ++q) {
            const int col = (qn0 + q) * 16 + lrow;
            v4f lo, hi;
            lo.x = acc[q][0] + br[0]; lo.y = acc[q][1] + br[1];
            lo.z = acc[q][2] + br[2]; lo.w = acc[q][3] + br[3];
            hi.x = acc[q][4] + br[4]; hi.y = acc[q][5] + br[5];
            hi.z = acc[q][6] + br[6]; hi.w = acc[q][7] + br[7];
            *(v4f*)&sA[col * SK + krow]     = lo;
            *(v4f*)&sA[col * SK + krow + 4] = hi;
        }
        __syncthreads();

        // ---- softmax over K per pixel: fully register-resident ----
        if (tid < PT) {
            const int base = tid * SK;
            v4f r[16];
            float m = -3.402823466e38f;
            #pragma unroll
            for (int j = 0; j < 16; ++j) {
                r[j] = *(const v4f*)&sA[base + 4 * j];
                m = fmaxf(m, fmaxf(fmaxf(r[j].x, r[j].y), fmaxf(r[j].z, r[j].w)));
            }
            float sum = 0.0f;
            #pragma unroll
            for (int j = 0; j < 16; ++j) {
                v4f e;
                e.x = __expf(r[j].x - m); e.y = __expf(r[j].y - m);
                e.z = __expf(r[j].z - m); e.w = __expf(r[j].w - m);
                sum += (e.x + e.y) + (e.z + e.w);
                r[j] = e;
            }
            const float inv = 1.0f / sum;
            #pragma unroll
            for (int j = 0; j < 16; ++j)
                *(v4f*)&sA[base + 4 * j] = r[j] * inv;
        }
        __syncthreads();
        if (tid < KK) {   // running column sums of the assignment matrix
            float s = 0.0f;
            for (int p = 0; p < PT; ++p) s += sA[p * SK + tid];
            sAsum[tid] += s;
        }

        // ---- GEMM2: vacc[k][c] += sum_p a[k][p] * xn[c][p] ----
        {
            const int kb = km * 16 + lrow;
            for (int p0 = 0; p0 < PT; p0 += 4) {
                v2f a;
                a.x = sA[(p0 + lkk) * SK + kb];
                a.y = sA[(p0 + lkk + 1) * SK + kb];
                #pragma unroll
                for (int q = 0; q < 4; ++q) {
                    const int ccol = (qn0 + q) * 16 + lrow;
                    v2f b;
                    b.x = bx[ccol * SX + p0 + lkk];
                    b.y = bx[ccol * SX + p0 + lkk + 1];
                    vacc[q] = wmma_f32(a, b, vacc[q]);
                }
            }
        }
        // no trailing barrier: next iteration's wait+barrier provides separation
    }

    // ---- flush per-block partials (4 chunks per image share an accumulator) ----
    float* vg = vlad_acc + (size_t)n * KK * CC;
    #pragma unroll
    for (int q = 0; q < 4; ++q) {
        const int ccol = (qn0 + q) * 16 + lrow;
        #pragma unroll
        for (int i = 0; i < 8; ++i)
            atomicAdd(&vg[(krow + i) * CC + ccol], vacc[q][i]);
    }
    __syncthreads();      // sAsum final values
    if (tid < KK) atomicAdd(&asum_acc[n * KK + tid], sAsum[tid]);
}

__global__ __launch_bounds__(256)
void netvlad_finalize(const float* __restrict__ vlad_acc,
                      const float* __restrict__ asum_acc,
                      const float* __restrict__ centroids,
                      float* __restrict__ out)
{
    __shared__ float vl[KK * CC];
    __shared__ float red[9];
    const int tid  = threadIdx.x;
    const int n    = blockIdx.x;
    const int wv   = tid >> 5;
    const int lane = tid & 31;
    const float* vg = vlad_acc + (size_t)n * KK * CC;

    // centroid subtraction + intra-normalization (per k over C)
    for (int k = wv; k < KK; k += 8) {
        const float asv = asum_acc[n * KK + k];
        float vals[4];
        float ss = 0.0f;
        #pragma unroll
        for (int j = 0; j < 4; ++j) {
            const int c = lane + j * 32;
            float v = vg[k * CC + c] - asv * centroids[k * CC + c];
            vals[j] = v;
            ss += v * v;
        }
        #pragma unroll
        for (int off = 16; off >= 1; off >>= 1) ss += __shfl_xor(ss, off, 32);
        const float inv = 1.0f / fmaxf(sqrtf(ss), 1e-12f);
        #pragma unroll
        for (int j = 0; j < 4; ++j) vl[k * CC + lane + j * 32] = vals[j] * inv;
    }
    __syncthreads();

    // global L2 normalization over K*C
    float gs = 0.0f;
    for (int i = tid; i < KK * CC; i += 256) gs += vl[i] * vl[i];
    #pragma unroll
    for (int off = 16; off >= 1; off >>= 1) gs += __shfl_xor(gs, off, 32);
    if (lane == 0) red[wv] = gs;
    __syncthreads();
    if (tid == 0) {
        float t = 0.0f;
        for (int w = 0; w < 8; ++w) t += red[w];
        red[8] = 1.0f / fmaxf(sqrtf(t), 1e-12f);
    }
    __syncthreads();
    const float ginv = red[8];
    for (int i = tid; i < KK * CC; i += 256)
        out[(size_t)n * KK * CC + i] = vl[i] * ginv;
}

extern "C" void kernel_launch(void* const* d_in, const int* in_sizes, int n_in,
                              void* d_out, int out_size, void* d_ws, size_t ws_size,
                              hipStream_t stream) {
    (void)in_sizes; (void)n_in; (void)out_size; (void)ws_size;
    const float* x      = (const float*)d_in[0];
    const float* conv_w = (const float*)d_in[1];
    const float* conv_b = (const float*)d_in[2];
    const float* cents  = (const float*)d_in[3];
    float* out = (float*)d_out;

    float* vlad_acc = (float*)d_ws;                       // N*K*C floats
    float* asum_acc = vlad_acc + (size_t)NN * KK * CC;    // N*K floats

    (void)hipMemsetAsync(d_ws, 0, ((size_t)NN * KK * CC + (size_t)NN * KK) * sizeof(float), stream);
    netvlad_main<<<NN * NCHUNK, 256, 0, stream>>>(x, conv_w, conv_b, vlad_acc, asum_acc);
    netvlad_finalize<<<NN, 256, 0, stream>>>(vlad_acc, asum_acc, cents, out);
}